// GraphPerformanceTrace_81698867904551
// MI455X (gfx1250) — compile-verified
//
#include <hip/hip_runtime.h>
#include <hip/hip_bf16.h>
#include <math.h>

// Problem constants (from reference)
constexpr int kN  = 512;   // nodes
constexpr int kBS = 64;    // batch
constexpr int kT  = 200;   // time steps
constexpr int kD  = 32;    // feature dim
constexpr float kALPHA = 0.1f;
constexpr float kEPS   = 1e-6f;

typedef __attribute__((ext_vector_type(2))) float v2f;
typedef __attribute__((ext_vector_type(8))) float v8f;

// ---------------------------------------------------------------------------
// Phase 1/2: batched fp32 GEMM with V_WMMA_F32_16X16X4_F32.
// Computes C[b] = A[b] @ B[b]  (all 512x512 row-major, batched over z grid).
// If fuse==1, stores  E1*inv_n + E2*inv_n2 + (A@B)*inv_n3  instead of raw A@B
// (used to build Msum = A1/n + A2/n^2 + A3/n^3 in one pass).
//
// Wave-level tiling: each wave owns a 16(row) x 64(col) strip of C:
//   4 fp32 accumulators (v8f each), A fragment reused across the 4 col-tiles.
// A-fragment layout (ISA 7.12.2, 32-bit A 16x4): lanes 0-15 hold K=k0,k0+1,
// lanes 16-31 hold K=k0+2,k0+3  -> one float2 load per lane per k-step.
// B-fragment: lanes 0-15 hold rows K=k0,k0+1 at col N=lane; lanes 16-31 hold
// K=k0+2,k0+3.  C/D layout: VGPR i -> M=i (lanes 0-15) / M=i+8 (lanes 16-31).
// ---------------------------------------------------------------------------
__global__ __launch_bounds__(256)
void gemm_wmma_f32(const float* __restrict__ A, const float* __restrict__ B,
                   float* __restrict__ C, int fuse,
                   const float* __restrict__ E1, const float* __restrict__ E2)
{
    const int lane = threadIdx.x & 31;
    const int wave = threadIdx.x >> 5;
    const int half = lane >> 4;      // 0: lanes 0-15, 1: lanes 16-31
    const int r    = lane & 15;

    const size_t boff = (size_t)blockIdx.z * kN * kN;
    const int row0 = blockIdx.y * 128 + wave * 16;   // 8 waves x 16 rows
    const int col0 = blockIdx.x * 64;                // 4 col-tiles of 16

    const float* Ab = A + boff;
    const float* Bb = B + boff;

    v8f acc0 = {}, acc1 = {}, acc2 = {}, acc3 = {};

    for (int k0 = 0; k0 < kN; k0 += 4) {
        // A fragment: float2 at A[row0+r, k0+2*half]
        const float* ap = Ab + (size_t)(row0 + r) * kN + (k0 + 2 * half);
        v2f a = *(const v2f*)ap;

        // B fragments: B[k0+2*half, col] and B[k0+2*half+1, col]
        const float* bp = Bb + (size_t)(k0 + 2 * half) * kN + col0 + r;
        v2f b0, b1, b2, b3;
        b0.x = bp[0];        b0.y = bp[kN];
        b1.x = bp[16];       b1.y = bp[kN + 16];
        b2.x = bp[32];       b2.y = bp[kN + 32];
        b3.x = bp[48];       b3.y = bp[kN + 48];

        acc0 = __builtin_amdgcn_wmma_f32_16x16x4_f32(false, a, false, b0, (short)0, acc0, false, false);
        acc1 = __builtin_amdgcn_wmma_f32_16x16x4_f32(false, a, false, b1, (short)0, acc1, false, false);
        acc2 = __builtin_amdgcn_wmma_f32_16x16x4_f32(false, a, false, b2, (short)0, acc2, false, false);
        acc3 = __builtin_amdgcn_wmma_f32_16x16x4_f32(false, a, false, b3, (short)0, acc3, false, false);
    }

    const float inv_n  = 1.0f / 512.0f;
    const float inv_n2 = 1.0f / (512.0f * 512.0f);
    const float inv_n3 = 1.0f / (512.0f * 512.0f * 512.0f);

    const int colr = col0 + r;
    #pragma unroll
    for (int i = 0; i < 8; ++i) {
        const int row = row0 + i + 8 * half;
        const size_t base = boff + (size_t)row * kN;
        float v0 = acc0[i], v1 = acc1[i], v2 = acc2[i], v3 = acc3[i];
        if (fuse) {
            v0 = E1[base + colr +  0] * inv_n + E2[base + colr +  0] * inv_n2 + v0 * inv_n3;
            v1 = E1[base + colr + 16] * inv_n + E2[base + colr + 16] * inv_n2 + v1 * inv_n3;
            v2 = E1[base + colr + 32] * inv_n + E2[base + colr + 32] * inv_n2 + v2 * inv_n3;
            v3 = E1[base + colr + 48] * inv_n + E2[base + colr + 48] * inv_n2 + v3 * inv_n3;
        }
        C[base + colr +  0] = v0;
        C[base + colr + 16] = v1;
        C[base + colr + 32] = v2;
        C[base + colr + 48] = v3;
    }
}

// ---------------------------------------------------------------------------
// Phase 3: the 199-step recurrence.  One workgroup (512 threads) per batch b;
// the whole scan runs inside the workgroup with carry state in LDS/registers.
// Per step, the dominant op is spatial[k] = sum_j lp[j] * Msum[b][j,k]
// (coalesced column-walk over the L2-resident 1 MB Msum slice).
// ---------------------------------------------------------------------------
__global__ __launch_bounds__(512, 1)
void scan_kernel(const int*   __restrict__ skills,
                 const int*   __restrict__ times,
                 const float* __restrict__ labels,
                 const float* __restrict__ ne,      // (N, 2D): [:,0:D]=send, [:,D:2D]=rec
                 const float* __restrict__ ub0,
                 const float* __restrict__ uf0,
                 const float* __restrict__ w1, const float* __restrict__ b1,
                 const float* __restrict__ w2, const float* __restrict__ b2,
                 const float* __restrict__ wf, const float* __restrict__ bf,
                 const float* __restrict__ wb, const float* __restrict__ bb,
                 const float* __restrict__ msum,    // (BS, N, N) precomputed
                 float*       __restrict__ preds)   // (T-1, BS)
{
    __shared__ float s_lp[kN];          // last_perf
    __shared__ float s_nlf[kN];         // node_last_forget
    __shared__ float s_ub[kD], s_uf[kD];
    __shared__ float s_h[kD], s_uu[kD];
    __shared__ float s_w1[kD * kD * 5];
    __shared__ float s_w2[kD * kD], s_wf[kD * kD], s_wb[kD * kD];
    __shared__ float s_b1[kD], s_b2[kD], s_bf[kD], s_bb[kD];

    const int b = blockIdx.x;
    const int k = threadIdx.x;          // node index 0..511

    // Cooperative weight load into LDS (weights are reused 199x).
    for (int i = k; i < kD * kD * 5; i += 512) s_w1[i] = w1[i];
    for (int i = k; i < kD * kD; i += 512) { s_w2[i] = w2[i]; s_wf[i] = wf[i]; s_wb[i] = wb[i]; }
    if (k < kD) {
        s_b1[k] = b1[k]; s_b2[k] = b2[k]; s_bf[k] = bf[k]; s_bb[k] = bb[k];
        s_ub[k] = ub0[b * kD + k];
        s_uf[k] = uf0[b * kD + k];
    }

    // Cache this thread's node-embedding row in registers (reused every step).
    float rs[kD], rr[kD];
    #pragma unroll
    for (int d = 0; d < kD; ++d) {
        rs[d] = ne[k * 2 * kD + d];          // node_send[k, d]
        rr[d] = ne[k * 2 * kD + kD + d];     // node_rec[k, d]
    }
    __syncthreads();

    // Initial carry: last_perf0 = ub0 @ rec^T ; nlf0 = uf0 @ send^T
    {
        float lp = 0.f, nf = 0.f;
        #pragma unroll
        for (int d = 0; d < kD; ++d) { lp += s_ub[d] * rr[d]; nf += s_uf[d] * rs[d]; }
        s_lp[k] = lp; s_nlf[k] = nf;
    }
    __syncthreads();

    const float* msum_b = msum + (size_t)b * kN * kN;
    const float inv_log5 = 1.0f / logf(5.0f);
    const float inv_3n   = 1.0f / (3.0f * 512.0f);

    for (int t = 0; t < kT - 1; ++t) {
        const int   t0 = times[b * kT + t];
        const int   t1 = times[b * kT + t + 1];
        const float delta_t = logf(fabsf((float)(t1 - t0)) + kEPS) * inv_log5;
        const float lab   = labels[b * kT + t + 1];
        const int   skill = skills[b * kT + t + 1];

        // spatial[k] = sum_j lp[j] * Msum[j, k]   (coalesced across the wave)
        float acc = 0.f;
        const float* mcol = msum_b + k;
        #pragma unroll 8
        for (int j = 0; j < kN; ++j) acc += s_lp[j] * mcol[(size_t)j * kN];

        // base / new node_last_forget from register-resident embedding row
        float basek = 0.f, nlfn = 0.f;
        #pragma unroll
        for (int d = 0; d < kD; ++d) { basek += s_ub[d] * rr[d]; nlfn += s_uf[d] * rs[d]; }

        const float temporal = __expf(-delta_t * kALPHA * s_nlf[k]) * s_lp[k];
        const float x  = basek + temporal + acc * inv_3n;
        const float cp = 1.0f / (1.0f + __expf(-x));   // sigmoid -> new last_perf

        // Small net (threads 0..31): h = relu(conv1(user_stack))
        float hval = 0.f;
        if (k < kD) {
            float a = s_b1[k];
            const float* wro = &s_w1[k * kD * 5];
            #pragma unroll 4
            for (int c = 0; c < kD; ++c) {
                const float sc = ne[skill * 2 * kD + c];
                const float rc = ne[skill * 2 * kD + kD + c];
                a += s_ub[c] * wro[c * 5 + 0]
                   + s_uf[c] * wro[c * 5 + 1]
                   + lab     * wro[c * 5 + 2]
                   + rc      * wro[c * 5 + 3]
                   + sc      * wro[c * 5 + 4];
            }
            hval = fmaxf(a, 0.0f);
        }
        __syncthreads();                   // all reads of old carry done

        s_lp[k]  = cp;
        s_nlf[k] = nlfn;
        if (k < kD) s_h[k] = hval;
        if (k == skill) preds[t * kBS + b] = cp;
        __syncthreads();

        if (k < kD) {                      // uu = h @ conv2^T + b2
            float u = s_b2[k];
            #pragma unroll 8
            for (int c = 0; c < kD; ++c) u += s_h[c] * s_w2[k * kD + c];
            s_uu[k] = u;
        }
        __syncthreads();

        if (k < kD) {                      // new_f/new_b = tanh(uu @ W^T + b)
            float f = s_bf[k], g = s_bb[k];
            #pragma unroll 8
            for (int c = 0; c < kD; ++c) {
                const float uc = s_uu[c];
                f += uc * s_wf[k * kD + c];
                g += uc * s_wb[k * kD + c];
            }
            s_uf[k] = tanhf(f);
            s_ub[k] = tanhf(g);
        }
        __syncthreads();
    }
}

// ---------------------------------------------------------------------------
extern "C" void kernel_launch(void* const* d_in, const int* in_sizes, int n_in,
                              void* d_out, int out_size, void* d_ws, size_t ws_size,
                              hipStream_t stream) {
    (void)in_sizes; (void)n_in; (void)out_size; (void)ws_size;

    const int*   skills = (const int*)  d_in[0];
    const int*   times  = (const int*)  d_in[1];
    const float* labels = (const float*)d_in[2];
    const float* adj    = (const float*)d_in[3];   // (1,BS,N,N) == (BS,N,N)
    const float* ne     = (const float*)d_in[4];   // (1,N,2D)
    const float* ub0    = (const float*)d_in[5];
    const float* uf0    = (const float*)d_in[6];
    const float* w1     = (const float*)d_in[7];
    const float* b1     = (const float*)d_in[8];
    const float* w2     = (const float*)d_in[9];
    const float* b2     = (const float*)d_in[10];
    const float* wf     = (const float*)d_in[11];
    const float* bf     = (const float*)d_in[12];
    const float* wb     = (const float*)d_in[13];
    const float* bb     = (const float*)d_in[14];
    float* preds = (float*)d_out;

    // Workspace: A2 (raw) then Msum, each BS*N*N floats (67 MB each, 134 MB total).
    float* a2ws   = (float*)d_ws;
    float* msumws = a2ws + (size_t)kBS * kN * kN;

    dim3 ggrid(kN / 64, kN / 128, kBS);   // (8, 4, 64), 8 waves/WG, 16x64 per wave

    // A2 = A1 @ A1   (raw)
    gemm_wmma_f32<<<ggrid, 256, 0, stream>>>(adj, adj, a2ws, 0, nullptr, nullptr);
    // Msum = A1/n + A2/n^2 + (A2 @ A1)/n^3   (fused epilogue)
    gemm_wmma_f32<<<ggrid, 256, 0, stream>>>(a2ws, adj, msumws, 1, adj, a2ws);

    // Full 199-step recurrence: one workgroup per batch, L2-resident Msum.
    scan_kernel<<<kBS, 512, 0, stream>>>(skills, times, labels, ne, ub0, uf0,
                                         w1, b1, w2, b2, wf, bf, wb, bb,
                                         msumws, preds);
}